// FieldAwareFactorizationMachine_49916109914363
// MI455X (gfx1250) — compile-verified
//
#include <hip/hip_runtime.h>

// FFM pairwise interactions for MI455X (gfx1250, wave32).
// One wave processes BATCH_PER_WAVE elements in a uniform loop; all lane-only
// addressing (triu inversion, LDS operand/store addresses) is hoisted out.
// Per element: gather -> wave-private LDS -> 12x v_wmma_f32_16x16x4_f32 Gram
//   -> Gram tiles to LDS (paired immediate-offset ds stores)
//   -> 276 coalesced b32 stores.

typedef __attribute__((ext_vector_type(2))) float v2f;
typedef __attribute__((ext_vector_type(8))) float v8f;

#define NUM_FIELDS 24
#define EMBED      16
#define VOCAB      100000
#define NPAIR      276           // 24*23/2
#define WAVES_PB   8
#define BATCH_PW   4             // batch elements per wave
#define XS_STRIDE  20            // floats: 80B rows, 16B-aligned b128 stores, conflict-free b64 reads
#define G_STRIDE   26            // floats: half-wave row groups hit disjoint bank ranges

// Row start of triu row i in the flat 276-vector: S(i) = i*(47-i)/2 (N=24).
__host__ __device__ constexpr int rowStart(int i) { return (i * (47 - i)) / 2; }

__global__ __launch_bounds__(256) void ffm_gram_wmma_kernel(
    const int* __restrict__ x,            // [batch, 24]
    const float* __restrict__ tables,     // [24, VOCAB, 16] f32
    float* __restrict__ out,              // [batch, 276] f32
    int batch)
{
    __shared__ float xs_s[WAVES_PB][32][XS_STRIDE];    // 20 KB
    __shared__ float gm_s[WAVES_PB][32][G_STRIDE];     // 26.6 KB

    const int lane = threadIdx.x & 31;
    const int wave = threadIdx.x >> 5;
    // wave-uniform first batch id, forced into an SGPR so loads/stores use saddr form
    const int b0 = __builtin_amdgcn_readfirstlane(
        (int)(blockIdx.x * (WAVES_PB * BATCH_PW) + wave * BATCH_PW));

    float (*xs)[XS_STRIDE] = xs_s[wave];
    float (*gm)[G_STRIDE]  = gm_s[wave];

    // ---- Hoisted lane-only state (loop-invariant) ----
    // WMMA A-operand addressing: lanes 0-15 hold M=0..15 K={4c,4c+1}; lanes 16-31 K={4c+2,4c+3}.
    const int r  = lane & 15;
    const int kh = (lane >> 4) << 1;

    // Gram tile store base (D layout: VGPR v -> M=v / v+8 per half-wave, N = lane&15).
    const int n0 = lane & 15;
    const int mA = (lane >> 4) << 3;
    float* g0 = &gm[mA][n0];

    // Triu inversion for the 9 output windows: t = 32u + lane. After full unroll the
    // window bounds are literals, so row index = constant base + a few (t >= literal) bumps.
    const float* gsrc[9];
#pragma unroll
    for (int u = 0; u < 9; ++u) {
        const int t  = u * 32 + lane;
        const int lo = u * 32, hi = u * 32 + 31;
        int i = 0;
#pragma unroll
        for (int rr = 1; rr <= 22; ++rr) {
            const int S = rowStart(rr);
            if (S <= lo)      i += 1;                 // folds to a constant
            else if (S <= hi) i += (t >= S) ? 1 : 0;  // 1 cmp+add per in-window boundary
        }
        const int s = (i * (47 - i)) >> 1;
        const int j = t - s + i + 1;                  // clamped lanes (u==8,t>=276) never loaded
        gsrc[u] = &gm[i < 23 ? i : 22][j < 32 ? j : 0];
    }

    // Zero the xs pad rows 24-31 once (never overwritten by the gather).
    if (lane >= NUM_FIELDS) {
        const float4 z = make_float4(0.f, 0.f, 0.f, 0.f);
        float4* dst = (float4*)&xs[lane][0];
        dst[0] = z; dst[1] = z; dst[2] = z; dst[3] = z;
    }

    // ---- Uniform batch loop (no barriers; LDS tile is wave-private, DS is in-order) ----
    for (int e = 0; e < BATCH_PW; ++e) {
        const int b = b0 + e;                // wave-uniform
        if (b >= batch) break;               // uniform exit

        // Gather: lane f pulls its 64B embedding row.
        if (lane < NUM_FIELDS) {
            const unsigned idx = (unsigned)x[b * NUM_FIELDS + lane];
            // 32-bit byte offset (max ~154MB < 2^31) off the uniform table base
            const unsigned byteoff =
                ((unsigned)lane * (unsigned)(VOCAB * EMBED) + idx * EMBED) * 4u;
            const float4* __restrict__ src = (const float4*)((const char*)tables + byteoff);
            float4* dst = (float4*)&xs[lane][0];
            dst[0] = src[0]; dst[1] = src[1]; dst[2] = src[2]; dst[3] = src[3];
        }

        // WMMA operands (B operand for Gram == A operand with matching base).
        v2f aLo[4], aHi[4];
#pragma unroll
        for (int c = 0; c < 4; ++c) {
            aLo[c] = *(const v2f*)&xs[r][4 * c + kh];        // fields 0..15
            aHi[c] = *(const v2f*)&xs[16 + r][4 * c + kh];   // fields 16..23 + zero pad
        }

        v8f d00 = {}; v8f d01 = {}; v8f d11 = {};
#pragma unroll
        for (int c = 0; c < 4; ++c) {   // K=16 in four x4 steps; EXEC all-ones here
            d00 = __builtin_amdgcn_wmma_f32_16x16x4_f32(false, aLo[c], false, aLo[c],
                                                        (short)0, d00, false, false);
            d01 = __builtin_amdgcn_wmma_f32_16x16x4_f32(false, aLo[c], false, aHi[c],
                                                        (short)0, d01, false, false);
            d11 = __builtin_amdgcn_wmma_f32_16x16x4_f32(false, aHi[c], false, aHi[c],
                                                        (short)0, d11, false, false);
        }

        // Gram tiles -> LDS: one base address + immediate offsets (pairs into ds_store_2addr).
#pragma unroll
        for (int v = 0; v < 8; ++v) {
            g0[v * G_STRIDE]             = d00[v];   // rows 0-15,  cols 0-15
            g0[v * G_STRIDE + 16]        = d01[v];   // rows 0-15,  cols 16-31
            g0[(16 + v) * G_STRIDE + 16] = d11[v];   // rows 16-31, cols 16-31
        }

        // 276 coalesced output stores from the hoisted gram addresses.
        float* __restrict__ ob = out + (size_t)b * NPAIR;
#pragma unroll
        for (int u = 0; u < 9; ++u) {
            const int t = u * 32 + lane;
            if (u < 8) {
                ob[t] = *gsrc[u];
            } else if (t < NPAIR) {          // only the last window predicates (lane<20)
                ob[t] = *gsrc[u];
            }
        }
    }
}

extern "C" void kernel_launch(void* const* d_in, const int* in_sizes, int n_in,
                              void* d_out, int out_size, void* d_ws, size_t ws_size,
                              hipStream_t stream) {
    const int*   x      = (const int*)d_in[0];     // [batch, 24]
    const float* tables = (const float*)d_in[1];   // [24, VOCAB, 16]
    float*       out    = (float*)d_out;           // [batch, 276]

    const int batch = in_sizes[0] / NUM_FIELDS;
    const int per_block = WAVES_PB * BATCH_PW;
    dim3 grid((batch + per_block - 1) / per_block);
    ffm_gram_wmma_kernel<<<grid, 256, 0, stream>>>(x, tables, out, batch);
}